// GCNConv_5111011083065
// MI455X (gfx1250) — compile-verified
//
#include <hip/hip_runtime.h>
#include <stdint.h>
#include <stddef.h>

#define D_FEAT 64
#define CHUNK 128              // edges per wave
#define GROUP 8                // gathers issued together (MLP per wave)
#define WAVES_PER_BLOCK 8
#define BLOCK_THREADS (WAVES_PER_BLOCK * 32)

// ---------------- CDNA5 async global->LDS copy (16B per lane) ----------------
// Probe result (round 1): builtin's param 0 is `int4 addrspace(1)*`.
typedef int v4i_ __attribute__((vector_size(16)));
typedef __attribute__((address_space(1))) v4i_* as1_v4i;
typedef __attribute__((address_space(3))) v4i_* as3_v4i;

__device__ __forceinline__ void async_cp16(void* lds, const void* g) {
#if __has_builtin(__builtin_amdgcn_global_load_async_to_lds_b128)
  __builtin_amdgcn_global_load_async_to_lds_b128(
      (as1_v4i)g, (as3_v4i)lds, 0, 0);
#else
  asm volatile("global_load_async_to_lds_b128 %0, %1, off"
               :
               : "v"(lds), "v"(g)
               : "memory");
#endif
}

__device__ __forceinline__ void wait_async_zero() {
#if __has_builtin(__builtin_amdgcn_s_wait_asynccnt)
  __builtin_amdgcn_s_wait_asynccnt(0);
  asm volatile("" ::: "memory");   // keep LDS reads below the wait
#else
  asm volatile("s_wait_asynccnt 0" ::: "memory");
#endif
}

// ---------------- segment flush: atomic add of a lane's float2 ----------------
__device__ __forceinline__ void flush_seg(float* __restrict__ out, int node,
                                          int lane, float2 acc) {
  float* p = out + (size_t)node * D_FEAT + (lane << 1);
  unsafeAtomicAdd(p,     acc.x);   // lowers to global_atomic_add_f32
  unsafeAtomicAdd(p + 1, acc.y);
}

// ---------------- main gather/scale/segment-scatter kernel ----------------
__global__ void __launch_bounds__(BLOCK_THREADS)
gcn_gather_scatter(const float* __restrict__ feat,
                   const float* __restrict__ ew,
                   const long long* __restrict__ esrc,
                   const long long* __restrict__ edst,
                   float* __restrict__ out,
                   int n_edges) {
  __shared__ long long s_src[WAVES_PER_BLOCK][CHUNK];   // 1 KB per wave
  __shared__ long long s_dst[WAVES_PER_BLOCK][CHUNK];   // 1 KB per wave
  __shared__ float     s_w  [WAVES_PER_BLOCK][CHUNK];   // 0.5 KB per wave

  const int lane = threadIdx.x & 31;
  const int wid  = threadIdx.x >> 5;

  const long long wave_id = (long long)blockIdx.x * WAVES_PER_BLOCK + wid;
  const long long e0 = wave_id * CHUNK;
  if (e0 >= n_edges) return;
  const int cn = (int)(((long long)n_edges - e0) < CHUNK ? ((long long)n_edges - e0)
                                                         : (long long)CHUNK);

  // ---- stage edge metadata into LDS via async copies (per-lane 16B) ----
  {
    const long long lim8 = (long long)n_edges * 8 - 16;  // int64 arrays
    const long long lim4 = (long long)n_edges * 4 - 16;  // float array
    const long long b8 = e0 * 8;
    const long long b4 = e0 * 4;
    char* lsrc = (char*)&s_src[wid][0];
    char* ldst = (char*)&s_dst[wid][0];
    char* lw   = (char*)&s_w[wid][0];
    const char* gsrc = (const char*)esrc;
    const char* gdst = (const char*)edst;
    const char* gw   = (const char*)ew;

    long long o;
    o = b8 + lane * 16;        if (o > lim8) o = lim8;
    async_cp16(lsrc + lane * 16,        gsrc + o);
    o = b8 + 512 + lane * 16;  if (o > lim8) o = lim8;
    async_cp16(lsrc + 512 + lane * 16,  gsrc + o);
    o = b8 + lane * 16;        if (o > lim8) o = lim8;
    async_cp16(ldst + lane * 16,        gdst + o);
    o = b8 + 512 + lane * 16;  if (o > lim8) o = lim8;
    async_cp16(ldst + 512 + lane * 16,  gdst + o);
    o = b4 + lane * 16;        if (o > lim4) o = lim4;
    async_cp16(lw + lane * 16,          gw + o);
  }
  wait_async_zero();

  // ---- process chunk: register segment accumulation over sorted dst ----
  const int* src32 = (const int*)&s_src[wid][0];  // low dword of each int64
  const int* dst32 = (const int*)&s_dst[wid][0];
  const float* wptr = &s_w[wid][0];

  int prev = -1;
  float2 acc; acc.x = 0.f; acc.y = 0.f;

  int e = 0;
  // Full groups: issue GROUP independent feature gathers before consuming any
  // -> GROUP-way memory-level parallelism per wave against L2.
  for (; e + GROUP <= cn; e += GROUP) {
    int s[GROUP];
#pragma unroll
    for (int i = 0; i < GROUP; ++i)
      s[i] = __builtin_amdgcn_readfirstlane(src32[2 * (e + i)]);

    float2 f[GROUP];
#pragma unroll
    for (int i = 0; i < GROUP; ++i)
      f[i] = *(const float2*)(feat + ((size_t)s[i] << 6) + (lane << 1));

    int d[GROUP]; float w[GROUP];
#pragma unroll
    for (int i = 0; i < GROUP; ++i) {
      d[i] = __builtin_amdgcn_readfirstlane(dst32[2 * (e + i)]);
      w[i] = wptr[e + i];
    }

#pragma unroll
    for (int i = 0; i < GROUP; ++i) {
      if (d[i] != prev) {              // scalar (wave-uniform) branch
        if (prev >= 0) flush_seg(out, prev, lane, acc);
        prev = d[i];
        acc.x = 0.f; acc.y = 0.f;
      }
      acc.x = fmaf(f[i].x, w[i], acc.x);
      acc.y = fmaf(f[i].y, w[i], acc.y);
    }
  }
  // tail
  for (; e < cn; ++e) {
    int s = __builtin_amdgcn_readfirstlane(src32[2 * e]);
    int d = __builtin_amdgcn_readfirstlane(dst32[2 * e]);
    float w = wptr[e];
    const float2 f = *(const float2*)(feat + ((size_t)s << 6) + (lane << 1));
    if (d != prev) {
      if (prev >= 0) flush_seg(out, prev, lane, acc);
      prev = d;
      acc.x = 0.f; acc.y = 0.f;
    }
    acc.x = fmaf(f.x, w, acc.x);
    acc.y = fmaf(f.y, w, acc.y);
  }
  if (prev >= 0) flush_seg(out, prev, lane, acc);
}

// ---------------- zero the output (harness poisons d_out) ----------------
__global__ void zero_kernel(float* __restrict__ out, int n) {
  int i = blockIdx.x * blockDim.x + threadIdx.x;
  int stride = gridDim.x * blockDim.x;
  float4* o4 = (float4*)out;
  int n4 = n >> 2;
  float4 z; z.x = 0.f; z.y = 0.f; z.z = 0.f; z.w = 0.f;
  for (int j = i; j < n4; j += stride) o4[j] = z;
  if (i == 0) {
    for (int j = (n4 << 2); j < n; ++j) out[j] = 0.f;
  }
}

extern "C" void kernel_launch(void* const* d_in, const int* in_sizes, int n_in,
                              void* d_out, int out_size, void* d_ws, size_t ws_size,
                              hipStream_t stream) {
  const float*     feat = (const float*)d_in[0];
  const float*     ew   = (const float*)d_in[1];
  const long long* esrc = (const long long*)d_in[2];
  const long long* edst = (const long long*)d_in[3];
  float* out = (float*)d_out;
  const int n_edges = in_sizes[1];

  // zero output
  {
    int n4 = out_size >> 2;
    int blocks = (n4 + 255) / 256;
    if (blocks < 1) blocks = 1;
    zero_kernel<<<blocks, 256, 0, stream>>>(out, out_size);
  }

  // main kernel: one wave per 128-edge chunk
  long long waves  = ((long long)n_edges + CHUNK - 1) / CHUNK;
  int       blocks = (int)((waves + WAVES_PER_BLOCK - 1) / WAVES_PER_BLOCK);
  gcn_gather_scatter<<<blocks, BLOCK_THREADS, 0, stream>>>(
      feat, ew, esrc, edst, out, n_edges);
}